// EdgeEnhancedSAGE_2697239462581
// MI455X (gfx1250) — compile-verified
//
#include <hip/hip_runtime.h>

typedef __attribute__((ext_vector_type(2))) float v2f;
typedef __attribute__((ext_vector_type(8))) float v8f;

constexpr int N_NODES = 100000;
constexpr int N_EDGES = 1600000;
constexpr int IN_F    = 48;
constexpr int EDGE_F  = 32;
constexpr int HID_F   = 128;
constexpr int OUT_F   = 64;

// ---------------------------------------------------------------------------
// Kernel 1: edge encoder + message + scatter-sum into h[:,48:96], degree count
//   e = relu(efeat @ We + be); msg = nfeat[src] * e; h2[dst] += msg; deg[dst]++
// ---------------------------------------------------------------------------
__global__ void __launch_bounds__(256)
k_edge_encode(const float* __restrict__ efeat, const float* __restrict__ nfeat,
              const int* __restrict__ src, const int* __restrict__ dst,
              const float* __restrict__ We, const float* __restrict__ be,
              float* __restrict__ h, float* __restrict__ deg)
{
    __shared__ float sWe[EDGE_F * IN_F];   // 6 KB, broadcast-read (no bank conflicts)
    __shared__ float sbe[IN_F];
    for (int i = threadIdx.x; i < EDGE_F * IN_F; i += blockDim.x) sWe[i] = We[i];
    if (threadIdx.x < IN_F) sbe[threadIdx.x] = be[threadIdx.x];
    __syncthreads();

    const int e = blockIdx.x * blockDim.x + threadIdx.x;
    if (e >= N_EDGES) return;

    float ef[EDGE_F];
    const float4* ef4 = reinterpret_cast<const float4*>(efeat + (size_t)e * EDGE_F);
    #pragma unroll
    for (int k = 0; k < EDGE_F / 4; ++k) {
        float4 v = ef4[k];
        ef[4*k+0] = v.x; ef[4*k+1] = v.y; ef[4*k+2] = v.z; ef[4*k+3] = v.w;
    }

    const int s = src[e], d = dst[e];
    atomicAdd(&deg[d], 1.0f);

    const float* nf = nfeat + (size_t)s * IN_F;
    float*       hd = h + (size_t)d * 96 + IN_F;

    #pragma unroll 4
    for (int j = 0; j < IN_F; ++j) {
        float acc = sbe[j];
        #pragma unroll
        for (int k = 0; k < EDGE_F; ++k) acc = fmaf(ef[k], sWe[k * IN_F + j], acc);
        acc = fmaxf(acc, 0.0f);               // relu
        atomicAdd(&hd[j], nf[j] * acc);
    }
}

// ---------------------------------------------------------------------------
// Kernel 2: h[:,0:48] = nfeat ; h[:,48:96] /= max(deg,1)
// ---------------------------------------------------------------------------
__global__ void __launch_bounds__(256)
k_finalize_h(const float* __restrict__ nfeat, const float* __restrict__ deg,
             float* __restrict__ h)
{
    const int t = blockIdx.x * blockDim.x + threadIdx.x;
    if (t >= N_NODES * IN_F) return;
    const int i = t / IN_F, j = t % IN_F;
    const float r = 1.0f / fmaxf(deg[i], 1.0f);
    h[(size_t)i * 96 + j] = nfeat[(size_t)i * IN_F + j];
    h[(size_t)i * 96 + IN_F + j] *= r;
}

// ---------------------------------------------------------------------------
// Scatter-add: one thread per (edge, column). Lanes cover consecutive columns
// of one edge -> coalesced gather rows + contiguous atomic cachelines (L2-hot).
// ---------------------------------------------------------------------------
template <int SRC_STRIDE, int COLS, int DST_STRIDE>
__global__ void __launch_bounds__(256)
k_scatter(const float* __restrict__ A, const int* __restrict__ src,
          const int* __restrict__ dst, float* __restrict__ B)
{
    const long long t = (long long)blockIdx.x * blockDim.x + threadIdx.x;
    if (t >= (long long)N_EDGES * COLS) return;
    const int e = (int)(t / COLS);
    const int j = (int)(t % COLS);
    atomicAdd(&B[(size_t)dst[e] * DST_STRIDE + j], A[(size_t)src[e] * SRC_STRIDE + j]);
}

template <int COLS, int STRIDE>
__global__ void __launch_bounds__(256)
k_normalize(float* __restrict__ buf, const float* __restrict__ deg)
{
    const int t = blockIdx.x * blockDim.x + threadIdx.x;
    if (t >= N_NODES * COLS) return;
    const int i = t / COLS, j = t % COLS;
    buf[(size_t)i * STRIDE + j] *= 1.0f / fmaxf(deg[i], 1.0f);
}

// ---------------------------------------------------------------------------
// Weight prep: WT[n*K + k] = concat(Ws, Wn) transposed, so a B-fragment
// (rows k0,k0+1 of column n) is ONE contiguous b64 load, with a wave-uniform
// base pointer (no divergent W1s/W1n selects in the GEMM hot loop).
// ---------------------------------------------------------------------------
template <int NC, int K, int KS>
__global__ void __launch_bounds__(256)
k_prep_wt(const float* __restrict__ Ws, const float* __restrict__ Wn,
          float* __restrict__ WT)
{
    const int t = blockIdx.x * blockDim.x + threadIdx.x;
    if (t >= NC * K) return;
    const int n = t / K, k = t % K;
    WT[t] = (k < KS) ? Ws[(size_t)k * NC + n] : Wn[(size_t)(k - KS) * NC + n];
}

// ---------------------------------------------------------------------------
// GEMM1 (fp32 WMMA): h1 = relu([h | hn1m] (100000x192) @ WT1^T (192x128) + b1)
// 8 waves/block; each wave: one 16-col tile x TWO 16-row tiles (2 WMMA per
// B-fragment load). Block covers 32 rows x 128 cols.
// ---------------------------------------------------------------------------
__global__ void __launch_bounds__(256)
k_gemm1(const float* __restrict__ h, const float* __restrict__ hn,
        const float* __restrict__ WT,   // [HID_F][192] transposed concat weights
        const float* __restrict__ b1, float* __restrict__ h1)
{
    constexpr int K = 192, NC = HID_F, LDA = 194;   // 194 % 64 == 2 -> conflict-free
    __shared__ float As[32 * LDA];
    const int base = blockIdx.x * 32;

    for (int idx = threadIdx.x; idx < 32 * K; idx += 256) {
        const int r = idx / K, c = idx % K;
        As[r * LDA + c] = (c < 96) ? h[(size_t)(base + r) * 96 + c]
                                   : hn[(size_t)(base + r) * 128 + (c - 96)];
    }
    __syncthreads();

    const int wave = threadIdx.x >> 5;
    const int lane = threadIdx.x & 31;
    const int rl   = lane & 15;      // row (A) / col (B) within tile
    const int hi   = lane >> 4;      // k-pair select per WMMA f32 layout
    const int col  = wave * 16 + rl;

    const float* wtc = WT + (size_t)col * K;         // per-lane B column, contiguous K
    const float* a0p = &As[rl * LDA];                // M-tile 0 row
    const float* a1p = &As[(16 + rl) * LDA];         // M-tile 1 row
    __builtin_prefetch(wtc, 0, 1);                   // global_prefetch_b8

    v8f c0 = {}, c1 = {};
    #pragma unroll 4
    for (int kk = 0; kk < K / 4; ++kk) {
        const int k0 = kk * 4 + 2 * hi;
        const v2f b  = *reinterpret_cast<const v2f*>(wtc + k0);  // global_load_b64
        const v2f a0 = *reinterpret_cast<const v2f*>(a0p + k0);  // ds_load_b64
        const v2f a1 = *reinterpret_cast<const v2f*>(a1p + k0);  // ds_load_b64
        c0 = __builtin_amdgcn_wmma_f32_16x16x4_f32(false, a0, false, b, (short)0, c0, false, false);
        c1 = __builtin_amdgcn_wmma_f32_16x16x4_f32(false, a1, false, b, (short)0, c1, false, false);
    }

    const float bias = b1[col];
    #pragma unroll
    for (int r = 0; r < 8; ++r) {                    // C layout: VGPR r -> M=r / M=r+8
        const int gr0 = base + r + 8 * hi;
        const int gr1 = base + 16 + r + 8 * hi;
        h1[(size_t)gr0 * NC + col] = fmaxf(c0[r] + bias, 0.0f);
        h1[(size_t)gr1 * NC + col] = fmaxf(c1[r] + bias, 0.0f);
    }
}

// ---------------------------------------------------------------------------
// GEMM2 (fp32 WMMA): out = [h1 | hn2m] (100000x256) @ WT2^T (256x64) + b2
// 4 waves/block (128 thr); each wave: one 16-col tile x TWO 16-row tiles.
// Block covers 32 rows x 64 cols.
// ---------------------------------------------------------------------------
__global__ void __launch_bounds__(128)
k_gemm2(const float* __restrict__ h1, const float* __restrict__ hn,
        const float* __restrict__ WT,   // [OUT_F][256] transposed concat weights
        const float* __restrict__ b2, float* __restrict__ out)
{
    constexpr int K = 256, NC = OUT_F, LDA = 258;    // 258 % 64 == 2 -> conflict-free
    __shared__ float As[32 * LDA];                   // ~33 KB
    const int base = blockIdx.x * 32;

    for (int idx = threadIdx.x; idx < 32 * K; idx += 128) {
        const int r = idx >> 8, cc = idx & 255;
        As[r * LDA + cc] = (cc < 128) ? h1[(size_t)(base + r) * 128 + cc]
                                      : hn[(size_t)(base + r) * 128 + (cc - 128)];
    }
    __syncthreads();

    const int wave = threadIdx.x >> 5;
    const int lane = threadIdx.x & 31;
    const int rl   = lane & 15;
    const int hi   = lane >> 4;
    const int col  = wave * 16 + rl;

    const float* wtc = WT + (size_t)col * K;
    const float* a0p = &As[rl * LDA];
    const float* a1p = &As[(16 + rl) * LDA];
    __builtin_prefetch(wtc, 0, 1);

    v8f c0 = {}, c1 = {};
    #pragma unroll 4
    for (int kk = 0; kk < K / 4; ++kk) {
        const int k0 = kk * 4 + 2 * hi;
        const v2f b  = *reinterpret_cast<const v2f*>(wtc + k0);
        const v2f a0 = *reinterpret_cast<const v2f*>(a0p + k0);
        const v2f a1 = *reinterpret_cast<const v2f*>(a1p + k0);
        c0 = __builtin_amdgcn_wmma_f32_16x16x4_f32(false, a0, false, b, (short)0, c0, false, false);
        c1 = __builtin_amdgcn_wmma_f32_16x16x4_f32(false, a1, false, b, (short)0, c1, false, false);
    }

    const float bias = b2[col];
    #pragma unroll
    for (int r = 0; r < 8; ++r) {
        const int gr0 = base + r + 8 * hi;
        const int gr1 = base + 16 + r + 8 * hi;
        out[(size_t)gr0 * NC + col] = c0[r] + bias;
        out[(size_t)gr1 * NC + col] = c1[r] + bias;
    }
}

// ---------------------------------------------------------------------------
extern "C" void kernel_launch(void* const* d_in, const int* in_sizes, int n_in,
                              void* d_out, int out_size, void* d_ws, size_t ws_size,
                              hipStream_t stream)
{
    const float* nfeat = (const float*)d_in[0];
    const float* efeat = (const float*)d_in[1];
    const int*   src   = (const int*)  d_in[2];
    const int*   dst   = (const int*)  d_in[3];
    const float* We    = (const float*)d_in[4];
    const float* be    = (const float*)d_in[5];
    const float* W1s   = (const float*)d_in[6];
    const float* W1n   = (const float*)d_in[7];
    const float* b1    = (const float*)d_in[8];
    const float* W2s   = (const float*)d_in[9];
    const float* W2n   = (const float*)d_in[10];
    const float* b2    = (const float*)d_in[11];
    float* out = (float*)d_out;

    // workspace: h[N*96] | hn[N*128] (reused both hops) | h1[N*128] | deg[N]
    //            | WT1[128*192] | WT2[64*256]
    float* h   = (float*)d_ws;
    float* hn  = h   + (size_t)N_NODES * 96;
    float* h1  = hn  + (size_t)N_NODES * 128;
    float* deg = h1  + (size_t)N_NODES * 128;
    float* WT1 = deg + (size_t)N_NODES;
    float* WT2 = WT1 + (size_t)HID_F * 192;

    hipMemsetAsync(h,   0, (size_t)N_NODES * 96  * sizeof(float), stream);
    hipMemsetAsync(hn,  0, (size_t)N_NODES * 128 * sizeof(float), stream);
    hipMemsetAsync(deg, 0, (size_t)N_NODES       * sizeof(float), stream);

    // Transposed concat weights (tiny, runs after the zero-fills on the same stream)
    k_prep_wt<HID_F, 192,  96><<<(HID_F * 192 + 255) / 256, 256, 0, stream>>>(W1s, W1n, WT1);
    k_prep_wt<OUT_F, 256, 128><<<(OUT_F * 256 + 255) / 256, 256, 0, stream>>>(W2s, W2n, WT2);

    // Hop 0: edge encoder + u_mul_e + scatter-sum + degree
    k_edge_encode<<<N_EDGES / 256, 256, 0, stream>>>(efeat, nfeat, src, dst, We, be, h, deg);
    k_finalize_h<<<(N_NODES * IN_F + 255) / 256, 256, 0, stream>>>(nfeat, deg, h);

    // Hop 1: mean-aggregate h over edges, SAGEConv1 via fp32 WMMA
    {
        unsigned int blocks = (unsigned int)(((long long)N_EDGES * 96 + 255) / 256);
        k_scatter<96, 96, 128><<<blocks, 256, 0, stream>>>(h, src, dst, hn);
    }
    k_normalize<96, 128><<<(N_NODES * 96 + 255) / 256, 256, 0, stream>>>(hn, deg);
    k_gemm1<<<N_NODES / 32, 256, 0, stream>>>(h, hn, WT1, b1, h1);

    // Hop 2: mean-aggregate h1, SAGEConv2 via fp32 WMMA
    hipMemsetAsync(hn, 0, (size_t)N_NODES * 128 * sizeof(float), stream);
    {
        unsigned int blocks = (unsigned int)(((long long)N_EDGES * 128 + 255) / 256);
        k_scatter<128, 128, 128><<<blocks, 256, 0, stream>>>(h1, src, dst, hn);
    }
    k_normalize<128, 128><<<(N_NODES * 128 + 255) / 256, 256, 0, stream>>>(hn, deg);
    k_gemm2<<<N_NODES / 32, 128, 0, stream>>>(h1, hn, WT2, b2, out);
}